// GraphSAGE_NET_56891136803141
// MI455X (gfx1250) — compile-verified
//
#include <hip/hip_runtime.h>
#include <hip/hip_bf16.h>

#define N_NODES 100000
#define N_EDGES 600000
#define F_IN    128
#define HID     128
#define CLS     16
#define NTILES  (N_NODES / 16)   // 6250 exactly -> no tail, EXEC all-1s for WMMA

typedef float v2f __attribute__((ext_vector_type(2)));
typedef float v8f __attribute__((ext_vector_type(8)));

// ---------------------------------------------------------------------------
// Edge degree: deg[dst] += 1.0  (hardware fp32 atomic)
// ---------------------------------------------------------------------------
__global__ __launch_bounds__(256) void deg_kernel(const int* __restrict__ dst,
                                                  float* __restrict__ deg) {
    int e = blockIdx.x * blockDim.x + threadIdx.x;
    if (e >= N_EDGES) return;
    unsafeAtomicAdd(&deg[dst[e]], 1.0f);
}

// ---------------------------------------------------------------------------
// Scatter-add: agg[dst][f] += feat[src][f]; one thread per (edge, feature)
// Coalesced across f (128 consecutive floats per edge).
// ---------------------------------------------------------------------------
__global__ __launch_bounds__(256) void scatter_kernel(const float* __restrict__ feat,
                                                      const int* __restrict__ src,
                                                      const int* __restrict__ dst,
                                                      float* __restrict__ agg) {
    long idx = (long)blockIdx.x * blockDim.x + threadIdx.x;
    if (idx >= (long)N_EDGES * 128) return;
    int e = (int)(idx >> 7);
    int f = (int)(idx & 127);
    int s = src[e];
    int d = dst[e];
    unsafeAtomicAdd(&agg[(long)d * 128 + f], feat[(long)s * 128 + f]);
}

// ---------------------------------------------------------------------------
// Layer 1: h = relu( (agg/deg) @ W_l + b + x @ W_r ),   K=128, Nout=128
// Block = 256 thr = 8 waves; block owns a 16-node tile staged in LDS,
// each wave owns one 16-wide output-column tile.
// v_wmma_f32_16x16x4_f32: A 16x4 (lanes 0-15: K=k0,k0+1 / lanes 16-31: K=k0+2,k0+3),
// B 4x16 mirrored, D 16x16 (VGPR r: lanes0-15 M=r, lanes16-31 M=r+8).
// ---------------------------------------------------------------------------
__global__ __launch_bounds__(256) void sage_layer1(const float* __restrict__ x,
                                                   const float* __restrict__ agg,
                                                   const float* __restrict__ deg,
                                                   const float* __restrict__ Wl,
                                                   const float* __restrict__ bias,
                                                   const float* __restrict__ Wr,
                                                   float* __restrict__ h) {
    __shared__ __align__(16) float As[16 * 132];   // agg/deg tile, padded stride
    __shared__ __align__(16) float Xs[16 * 132];   // x tile

    const int  tid      = threadIdx.x;
    const int  lane     = tid & 31;
    const int  wave     = tid >> 5;                // 0..7 -> output column tile
    const long nodeBase = (long)blockIdx.x * 16;

    // Cooperative stage: 2048 floats per array, float4 per thread x2
    #pragma unroll
    for (int j = 0; j < 2; ++j) {
        int  i    = tid * 8 + j * 4;               // 0..2047
        int  m    = i >> 7;                        // node row in tile
        int  k    = i & 127;
        long node = nodeBase + m;
        float inv = 1.0f / fmaxf(deg[node], 1.0f);
        float4 av = *(const float4*)(agg + node * 128 + k);
        float4 xv = *(const float4*)(x   + node * 128 + k);
        av.x *= inv; av.y *= inv; av.z *= inv; av.w *= inv;
        *(float4*)(&As[m * 132 + k]) = av;
        *(float4*)(&Xs[m * 132 + k]) = xv;
    }
    __syncthreads();

    const int m    = lane & 15;
    const int koff = (lane >> 4) * 2;
    const int n    = wave * 16 + (lane & 15);

    v8f c = {};
    #pragma unroll 4
    for (int k0 = 0; k0 < 128; k0 += 4) {
        v2f a, b;
        a.x = As[m * 132 + k0 + koff];
        a.y = As[m * 132 + k0 + koff + 1];
        b.x = Wl[(k0 + koff) * HID + n];
        b.y = Wl[(k0 + koff + 1) * HID + n];
        c = __builtin_amdgcn_wmma_f32_16x16x4_f32(false, a, false, b,
                                                  (short)0, c, false, false);
    }
    #pragma unroll 4
    for (int k0 = 0; k0 < 128; k0 += 4) {
        v2f a, b;
        a.x = Xs[m * 132 + k0 + koff];
        a.y = Xs[m * 132 + k0 + koff + 1];
        b.x = Wr[(k0 + koff) * HID + n];
        b.y = Wr[(k0 + koff + 1) * HID + n];
        c = __builtin_amdgcn_wmma_f32_16x16x4_f32(false, a, false, b,
                                                  (short)0, c, false, false);
    }

    const float bv = bias[n];
    #pragma unroll
    for (int r = 0; r < 8; ++r) {
        int   row = (lane < 16) ? r : (r + 8);
        float v   = fmaxf(c[r] + bv, 0.0f);        // bias + ReLU
        h[(nodeBase + row) * 128 + n] = v;
    }
}

// ---------------------------------------------------------------------------
// Layer 2: logits = (agg/deg) @ W2_l + b2 + h @ W2_r,   K=128, Nout=16
// One wave per 16-node tile (16 cols = full CLS). 8 waves / block.
// ---------------------------------------------------------------------------
__global__ __launch_bounds__(256) void sage_layer2(const float* __restrict__ hfeat,
                                                   const float* __restrict__ agg,
                                                   const float* __restrict__ deg,
                                                   const float* __restrict__ Wl,
                                                   const float* __restrict__ bias,
                                                   const float* __restrict__ Wr,
                                                   float* __restrict__ out) {
    const int  lane = threadIdx.x & 31;
    const int  wave = threadIdx.x >> 5;
    const long tile = (long)blockIdx.x * 8 + wave;
    if (tile >= NTILES) return;                    // wave-uniform exit

    const long nodeBase = tile * 16;
    const int  m    = lane & 15;
    const int  koff = (lane >> 4) * 2;
    const int  n    = lane & 15;
    const long nodeA = nodeBase + m;
    const float inv  = 1.0f / fmaxf(deg[nodeA], 1.0f);

    v8f c = {};
    #pragma unroll 4
    for (int k0 = 0; k0 < 128; k0 += 4) {
        v2f a, b;
        a.x = agg[nodeA * 128 + k0 + koff] * inv;
        a.y = agg[nodeA * 128 + k0 + koff + 1] * inv;
        b.x = Wl[(k0 + koff) * CLS + n];
        b.y = Wl[(k0 + koff + 1) * CLS + n];
        c = __builtin_amdgcn_wmma_f32_16x16x4_f32(false, a, false, b,
                                                  (short)0, c, false, false);
    }
    #pragma unroll 4
    for (int k0 = 0; k0 < 128; k0 += 4) {
        v2f a, b;
        a.x = hfeat[nodeA * 128 + k0 + koff];
        a.y = hfeat[nodeA * 128 + k0 + koff + 1];
        b.x = Wr[(k0 + koff) * CLS + n];
        b.y = Wr[(k0 + koff + 1) * CLS + n];
        c = __builtin_amdgcn_wmma_f32_16x16x4_f32(false, a, false, b,
                                                  (short)0, c, false, false);
    }

    const float bv = bias[n];
    #pragma unroll
    for (int r = 0; r < 8; ++r) {
        int row = (lane < 16) ? r : (r + 8);
        out[(nodeBase + row) * CLS + n] = c[r] + bv;
    }
}

// ---------------------------------------------------------------------------
// In-place log_softmax over 16 classes, one thread per node
// ---------------------------------------------------------------------------
__global__ __launch_bounds__(256) void logsoftmax_kernel(float* __restrict__ out) {
    int i = blockIdx.x * blockDim.x + threadIdx.x;
    if (i >= N_NODES) return;
    float v[CLS];
    float mx = -3.402823466e38f;
    #pragma unroll
    for (int j = 0; j < CLS; ++j) {
        v[j] = out[(long)i * CLS + j];
        mx   = fmaxf(mx, v[j]);
    }
    float s = 0.0f;
    #pragma unroll
    for (int j = 0; j < CLS; ++j) s += __expf(v[j] - mx);
    float lse = mx + __logf(s);
    #pragma unroll
    for (int j = 0; j < CLS; ++j) out[(long)i * CLS + j] = v[j] - lse;
}

// ---------------------------------------------------------------------------
// Launch
// ---------------------------------------------------------------------------
extern "C" void kernel_launch(void* const* d_in, const int* in_sizes, int n_in,
                              void* d_out, int out_size, void* d_ws, size_t ws_size,
                              hipStream_t stream) {
    const float* x    = (const float*)d_in[0];
    const int*   ei   = (const int*)d_in[1];      // harness contract: integer -> int32
    const float* W1l  = (const float*)d_in[2];
    const float* b1   = (const float*)d_in[3];
    const float* W1r  = (const float*)d_in[4];
    const float* W2l  = (const float*)d_in[5];
    const float* b2   = (const float*)d_in[6];
    const float* W2r  = (const float*)d_in[7];
    const int* src = ei;
    const int* dst = ei + N_EDGES;
    float* out = (float*)d_out;

    // Workspace layout (floats): deg | agg | h
    const long DEG_PAD = 100096;                  // N_NODES rounded up to 256
    float* deg = (float*)d_ws;
    float* agg = deg + DEG_PAD;
    float* h   = agg + (long)N_NODES * 128;

    // ---- Layer 1 ----
    hipMemsetAsync(deg, 0, (DEG_PAD + (long)N_NODES * 128) * sizeof(float), stream);
    deg_kernel<<<(N_EDGES + 255) / 256, 256, 0, stream>>>(dst, deg);
    {
        long total = (long)N_EDGES * 128;
        scatter_kernel<<<(int)((total + 255) / 256), 256, 0, stream>>>(x, src, dst, agg);
    }
    sage_layer1<<<NTILES, 256, 0, stream>>>(x, agg, deg, W1l, b1, W1r, h);

    // ---- Layer 2 ----
    hipMemsetAsync(agg, 0, (long)N_NODES * 128 * sizeof(float), stream);
    {
        long total = (long)N_EDGES * 128;
        scatter_kernel<<<(int)((total + 255) / 256), 256, 0, stream>>>(h, src, dst, agg);
    }
    sage_layer2<<<(NTILES + 7) / 8, 256, 0, stream>>>(h, agg, deg, W2l, b2, W2r, out);

    // ---- log_softmax ----
    logsoftmax_kernel<<<(N_NODES + 255) / 256, 256, 0, stream>>>(out);
}